// GAT_1133871366711
// MI455X (gfx1250) — compile-verified
//
#include <hip/hip_runtime.h>
#include <math.h>

typedef __attribute__((ext_vector_type(2))) float v2f;
typedef __attribute__((ext_vector_type(8))) float v8f;

#define KDIM 128      // all GEMMs in this net have K = 128
#define AST  132      // A LDS row stride (128 + 4 pad -> conflict-free column reads)
#define BST  68       // B LDS row stride (64 + 4 pad)

// Async global->LDS copy of 16 bytes per lane (CDNA5 ASYNCcnt path).
// VDST VGPR carries the per-lane LDS byte offset; low 32 bits of a flat
// LDS pointer are exactly the LDS offset (aperture base has zero low bits).
__device__ __forceinline__ void async_copy16(const float* gsrc, float* ldst) {
  unsigned loff = (unsigned)(unsigned long long)ldst;
  asm volatile("global_load_async_to_lds_b128 %0, %1, off"
               :: "v"(loff), "v"(gsrc)
               : "memory");
}
__device__ __forceinline__ void wait_async0() {
  asm volatile("s_wait_asynccnt 0" ::: "memory");
}

// ---------------------------------------------------------------------------
// GEMM: C[M,Ncols] = A[M,128] @ B[128,Ncols], fp32 via V_WMMA_F32_16X16X4_F32.
// Block = 256 threads = 8 waves (2 row-tiles x 4 col-tiles of 16x16).
// Tiles staged with GLOBAL_LOAD_ASYNC_TO_LDS_B128.
// ---------------------------------------------------------------------------
__global__ __launch_bounds__(256) void gemm128_wmma(const float* __restrict__ A,
                                                    const float* __restrict__ B,
                                                    float* __restrict__ C,
                                                    int M, int Ncols) {
  __shared__ float As[32 * AST];
  __shared__ float Bs[KDIM * BST];
  const int tid     = threadIdx.x;
  const int rowBase = blockIdx.x * 32;
  const int colBase = blockIdx.y * 64;

  // Stage A tile: 32 rows x 128 cols. Valid rows stream straight into LDS via
  // the async engine; OOB rows get zero-filled (disjoint LDS addrs -> no race).
  for (int i = tid; i < 32 * 32; i += 256) {
    int r = i >> 5;
    int c = (i & 31) << 2;
    int gr = rowBase + r;
    float* ldst = &As[r * AST + c];
    if (gr < M) {
      async_copy16(A + (size_t)gr * KDIM + c, ldst);
    } else {
      ldst[0] = 0.f; ldst[1] = 0.f; ldst[2] = 0.f; ldst[3] = 0.f;
    }
  }
  // Stage B tile: 128 rows x 64 cols (OOB cols -> 0)
  for (int i = tid; i < 128 * 16; i += 256) {
    int r = i >> 4;
    int c = (i & 15) << 2;
    int gc = colBase + c;
    float* ldst = &Bs[r * BST + c];
    if (gc + 3 < Ncols) {
      async_copy16(B + (size_t)r * Ncols + gc, ldst);
    } else {
      ldst[0] = (gc + 0 < Ncols) ? B[(size_t)r * Ncols + gc + 0] : 0.f;
      ldst[1] = (gc + 1 < Ncols) ? B[(size_t)r * Ncols + gc + 1] : 0.f;
      ldst[2] = (gc + 2 < Ncols) ? B[(size_t)r * Ncols + gc + 2] : 0.f;
      ldst[3] = (gc + 3 < Ncols) ? B[(size_t)r * Ncols + gc + 3] : 0.f;
    }
  }
  wait_async0();       // this wave's async LDS writes complete
  __syncthreads();     // all waves' tiles visible

  const int wave  = tid >> 5;
  const int lane  = tid & 31;
  const int wr    = (wave >> 2) << 4;   // 0 or 16 (tile row offset)
  const int wc    = (wave &  3) << 4;   // 0..48  (tile col offset)
  const int half  = lane >> 4;          // 0: lanes 0-15, 1: lanes 16-31
  const int l16   = lane & 15;

  v8f acc = {};
#pragma unroll
  for (int k = 0; k < KDIM; k += 4) {
    // A frag (16x4): lanes 0-15 hold K=k,k+1; lanes 16-31 hold K=k+2,k+3 (M = l16)
    const float* ap = &As[(wr + l16) * AST + k + 2 * half];
    v2f a; a.x = ap[0]; a.y = ap[1];
    // B frag (4x16): VGPR0 = row k(+2), VGPR1 = row k+1(+3); N = l16
    v2f b;
    b.x = Bs[(k + 2 * half + 0) * BST + wc + l16];
    b.y = Bs[(k + 2 * half + 1) * BST + wc + l16];
    acc = __builtin_amdgcn_wmma_f32_16x16x4_f32(false, a, false, b,
                                                (short)0, acc, false, false);
  }

  // D layout: VGPR v -> (M = v + 8*half, N = l16)
  const int col = colBase + wc + l16;
  if (col < Ncols) {
#pragma unroll
    for (int v = 0; v < 8; ++v) {
      int row = rowBase + wr + v + 8 * half;
      if (row < M) C[(size_t)row * Ncols + col] = acc[v];
    }
  }
}

// ---------------------------------------------------------------------------
// Attention coefficients: el[n,h] = sum_d feat[n,h,d]*al[h,d]; same for er.
// ---------------------------------------------------------------------------
__global__ void attn_coef(const float* __restrict__ feat,
                          const float* __restrict__ al,
                          const float* __restrict__ ar,
                          float* __restrict__ el, float* __restrict__ er,
                          int N, int H, int D) {
  int idx = blockIdx.x * blockDim.x + threadIdx.x;   // idx = n*H + h
  if (idx >= N * H) return;
  int h = idx % H;
  const float* f = feat + (size_t)idx * D;
  float sl = 0.f, sr = 0.f;
  for (int d = 0; d < D; ++d) {
    float x = f[d];
    sl = fmaf(x, al[h * D + d], sl);
    sr = fmaf(x, ar[h * D + d], sr);
  }
  el[idx] = sl;
  er[idx] = sr;
}

__global__ void fill_val(float* __restrict__ p, float v, int n) {
  int i = blockIdx.x * blockDim.x + threadIdx.x;
  if (i < n) p[i] = v;
}

__device__ inline void atomicMaxF(float* addr, float val) {
  unsigned int* u = reinterpret_cast<unsigned int*>(addr);
  unsigned int old = *u;
  while (__uint_as_float(old) < val) {
    unsigned int assumed = old;
    old = atomicCAS(u, assumed, __float_as_uint(val));
    if (old == assumed) break;
  }
}

// Edge logits: e = leaky_relu(el[src]+er[dst]); segment max over dst.
__global__ void edge_logit(const int* __restrict__ src, const int* __restrict__ dst,
                           const float* __restrict__ el, const float* __restrict__ er,
                           float* __restrict__ e, float* __restrict__ m,
                           int E, int H) {
  int idx = blockIdx.x * blockDim.x + threadIdx.x;   // idx = edge*H + h
  if (idx >= E * H) return;
  int ed = idx / H, h = idx - ed * H;
  float x = el[src[ed] * H + h] + er[dst[ed] * H + h];
  x = x > 0.f ? x : 0.2f * x;
  e[idx] = x;
  atomicMaxF(&m[dst[ed] * H + h], x);
}

// Nodes with no in-edges: m = -inf -> 0 (matches jnp.where(isfinite)).
__global__ void fix_m(float* __restrict__ m, int n) {
  int i = blockIdx.x * blockDim.x + threadIdx.x;
  if (i < n && !(m[i] > -__builtin_huge_valf())) m[i] = 0.f;
}

// ex = exp(e - m[dst]); den[dst] += ex
__global__ void edge_exp(const int* __restrict__ dst, float* __restrict__ e,
                         const float* __restrict__ m, float* __restrict__ den,
                         int E, int H) {
  int idx = blockIdx.x * blockDim.x + threadIdx.x;
  if (idx >= E * H) return;
  int ed = idx / H, h = idx - ed * H;
  float ex = __expf(e[idx] - m[dst[ed] * H + h]);
  e[idx] = ex;
  atomicAdd(&den[dst[ed] * H + h], ex);
}

// Message passing: out[dst] += feat[src] * alpha. One wave32 per edge.
__global__ __launch_bounds__(256) void scatter_msg(const int* __restrict__ src,
                                                   const int* __restrict__ dst,
                                                   const float* __restrict__ feat,
                                                   const float* __restrict__ ex,
                                                   const float* __restrict__ den,
                                                   float* __restrict__ out,
                                                   int E, int H, int D) {
  int wave = (blockIdx.x * blockDim.x + threadIdx.x) >> 5;
  int lane = threadIdx.x & 31;
  if (wave >= E) return;
  int s = src[wave], d = dst[wave];
  int HD = H * D;
  for (int el = lane; el < HD; el += 32) {
    int h = el / D;
    float alpha = ex[(size_t)wave * H + h] / (den[d * H + h] + 1e-9f);
    atomicAdd(&out[(size_t)d * HD + el], feat[(size_t)s * HD + el] * alpha);
  }
}

// out = elu(out + optional residual)
__global__ void elu_res(float* __restrict__ out, const float* __restrict__ res, int n) {
  int i = blockIdx.x * blockDim.x + threadIdx.x;
  if (i >= n) return;
  float x = out[i] + (res ? res[i] : 0.f);
  out[i] = x > 0.f ? x : (__expf(x) - 1.f);
}

// ---------------------------------------------------------------------------
// Host-side layer driver (all launches on `stream`, graph-capture safe).
// ---------------------------------------------------------------------------
static void gat_layer(const float* h, const float* W, const float* al, const float* ar,
                      const int* src, const int* dst,
                      float* feat, float* out,
                      float* el, float* er, float* m, float* den, float* ee,
                      int N, int E, int H, int D, bool zero_out, hipStream_t s) {
  const int HD = H * D;
  dim3 gg((N + 31) / 32, (HD + 63) / 64);
  gemm128_wmma<<<gg, 256, 0, s>>>(h, W, feat, N, HD);
  int nh = N * H;
  attn_coef<<<(nh + 255) / 256, 256, 0, s>>>(feat, al, ar, el, er, N, H, D);
  fill_val<<<(nh + 255) / 256, 256, 0, s>>>(m, -__builtin_huge_valf(), nh);
  fill_val<<<(nh + 255) / 256, 256, 0, s>>>(den, 0.f, nh);
  if (zero_out) {
    int no = N * HD;
    fill_val<<<(no + 255) / 256, 256, 0, s>>>(out, 0.f, no);
  }
  int eh = E * H;
  edge_logit<<<(eh + 255) / 256, 256, 0, s>>>(src, dst, el, er, ee, m, E, H);
  fix_m<<<(nh + 255) / 256, 256, 0, s>>>(m, nh);
  edge_exp<<<(eh + 255) / 256, 256, 0, s>>>(dst, ee, m, den, E, H);
  scatter_msg<<<(E + 7) / 8, 256, 0, s>>>(src, dst, feat, ee, den, out, E, H, D);
}

extern "C" void kernel_launch(void* const* d_in, const int* in_sizes, int n_in,
                              void* d_out, int out_size, void* d_ws, size_t ws_size,
                              hipStream_t stream) {
  const float* features = (const float*)d_in[0];
  const float* W0    = (const float*)d_in[1];
  const float* al0   = (const float*)d_in[2];
  const float* ar0   = (const float*)d_in[3];
  const float* W1    = (const float*)d_in[4];
  const float* al1   = (const float*)d_in[5];
  const float* ar1   = (const float*)d_in[6];
  const float* W2    = (const float*)d_in[7];
  const float* al2   = (const float*)d_in[8];
  const float* ar2   = (const float*)d_in[9];
  const float* resW2 = (const float*)d_in[10];
  const int*   src   = (const int*)d_in[11];
  const int*   dst   = (const int*)d_in[12];

  const int N = in_sizes[0] / 128;
  const int E = in_sizes[11];

  float* ws = (float*)d_ws;
  const size_t NF = (size_t)N * 128;
  float* feat = ws;                 // [N,128] projected features (per layer)
  float* h1   = ws + NF;            // [N,128] layer-0 output
  float* h2   = ws + 2 * NF;        // [N,128] layer-1 output
  float* el   = ws + 3 * NF;        // [N,4]
  float* er   = el + (size_t)N * 4;
  float* m    = er + (size_t)N * 4;
  float* den  = m  + (size_t)N * 4;
  float* ee   = den + (size_t)N * 4; // [E,4] edge logits / exp
  float* out  = (float*)d_out;       // [N,40]

  // ---- Layer 0: 4 heads x 32, no residual, ELU ----
  gat_layer(features, W0, al0, ar0, src, dst, feat, h1, el, er, m, den, ee,
            N, E, 4, 32, true, stream);
  {
    int n = N * 128;
    elu_res<<<(n + 255) / 256, 256, 0, stream>>>(h1, nullptr, n);
  }

  // ---- Layer 1: 4 heads x 32, identity residual, ELU ----
  gat_layer(h1, W1, al1, ar1, src, dst, feat, h2, el, er, m, den, ee,
            N, E, 4, 32, true, stream);
  {
    int n = N * 128;
    elu_res<<<(n + 255) / 256, 256, 0, stream>>>(h2, h1, n);
  }

  // ---- Layer 2: 1 head x 40, projected residual, mean over heads (H=1 -> id) ----
  // Seed d_out with the residual projection h2 @ resW2, then scatter-add on top.
  {
    dim3 gg((N + 31) / 32, (40 + 63) / 64);
    gemm128_wmma<<<gg, 256, 0, stream>>>(h2, resW2, out, N, 40);
  }
  gat_layer(h2, W2, al2, ar2, src, dst, feat, out, el, er, m, den, ee,
            N, E, 1, 40, false, stream);
}